// LogSumExpNegDist_64510408786521
// MI455X (gfx1250) — compile-verified
//
#include <hip/hip_runtime.h>
#include <hip/hip_bf16.h>
#include <stdint.h>

#define N_COLS 4096
#define BLOCK  256
#define CHUNKS 4          // 4 x b128 per thread = 16 floats
#define ELEMS  (CHUNKS * 4)

// ---- CDNA5 async global->LDS path (gfx1250), with safe fallbacks ----------
#if defined(__AMDGCN__)
  #if __has_builtin(__builtin_amdgcn_global_load_async_to_lds_b128)
    #define HAVE_ASYNC_LDS 1
  #endif
  #if __has_builtin(__builtin_amdgcn_s_wait_asynccnt)
    #define HAVE_WAIT_ASYNC 1
  #endif
#endif

// The async-to-LDS builtin takes pointers to 16-byte int vectors:
//   param0: v4i in AS1 (global / "__device__"), param1: v4i in AS3 (LDS)
typedef int v4i __attribute__((ext_vector_type(4)));
typedef __attribute__((address_space(1))) v4i* gbl_v4i_p;
typedef __attribute__((address_space(3))) v4i* lds_v4i_p;

// Generic->AS conversion via integer round-trip (generic shared ptr's low
// 32 bits are the LDS byte offset on amdgcn).
__device__ __forceinline__ gbl_v4i_p as_gbl_v4(const void* p) {
  return (gbl_v4i_p)(uintptr_t)p;
}
__device__ __forceinline__ lds_v4i_p as_lds_v4(const void* p) {
  return (lds_v4i_p)(uintptr_t)(uint32_t)(uintptr_t)p;
}

// ---- wave32 reductions ----------------------------------------------------
__device__ __forceinline__ float wave_sum(float v) {
#pragma unroll
  for (int o = 16; o > 0; o >>= 1) v += __shfl_xor(v, o, 32);
  return v;
}
__device__ __forceinline__ float wave_min(float v) {
#pragma unroll
  for (int o = 16; o > 0; o >>= 1) v = fminf(v, __shfl_xor(v, o, 32));
  return v;
}

__device__ __forceinline__ float block_sum(float v, float* red) {
  const int lane = threadIdx.x & 31;
  const int wv   = threadIdx.x >> 5;
  v = wave_sum(v);
  if (lane == 0) red[wv] = v;
  __syncthreads();
  if (wv == 0) {
    float t = (lane < (BLOCK / 32)) ? red[lane] : 0.0f;
    t = wave_sum(t);
    if (lane == 0) red[0] = t;
  }
  __syncthreads();
  float r = red[0];
  __syncthreads();
  return r;
}

__device__ __forceinline__ float block_min(float v, float* red) {
  const int lane = threadIdx.x & 31;
  const int wv   = threadIdx.x >> 5;
  v = wave_min(v);
  if (lane == 0) red[wv] = v;
  __syncthreads();
  if (wv == 0) {
    float t = (lane < (BLOCK / 32)) ? red[lane] : 3.0e38f;
    t = wave_min(t);
    if (lane == 0) red[0] = t;
  }
  __syncthreads();
  float r = red[0];
  __syncthreads();
  return r;
}

// ---- per-row logsumexp(-dist) kernel --------------------------------------
__global__ void __launch_bounds__(BLOCK)
row_lse_kernel(const float* __restrict__ x, float* __restrict__ row_lse, int M) {
  __shared__ __align__(16) float row[N_COLS];
  __shared__ float red[BLOCK / 32];

  const int m = blockIdx.x;
  const int t = threadIdx.x;
  const float* g = x + (size_t)m * N_COLS;

  // Stage the 16KB row into LDS via the CDNA5 async copy engine (ASYNCcnt
  // path: no VGPR staging, coalesced b128 per lane).
#if HAVE_ASYNC_LDS
#pragma unroll
  for (int k = 0; k < CHUNKS; ++k) {
    const int c = t + k * BLOCK;                 // 16-byte chunk index
    __builtin_amdgcn_global_load_async_to_lds_b128(
        as_gbl_v4(g + c * 4), as_lds_v4(&row[c * 4]), 0, 0);
  }
#if HAVE_WAIT_ASYNC
  __builtin_amdgcn_s_wait_asynccnt(0);
#else
  asm volatile("s_wait_asynccnt 0" ::: "memory");
#endif
#else
#pragma unroll
  for (int k = 0; k < CHUNKS; ++k) {
    const int c = t + k * BLOCK;
    ((float4*)row)[c] = ((const float4*)g)[c];
  }
#endif
  __syncthreads();

  // Pass 1: squared norm of the row.
  const float4* row4 = (const float4*)row;
  float4 vals[CHUNKS];
  float ss = 0.0f;
#pragma unroll
  for (int k = 0; k < CHUNKS; ++k) {
    float4 v = row4[t + k * BLOCK];
    vals[k] = v;
    ss += v.x * v.x + v.y * v.y + v.z * v.z + v.w * v.w;
  }
  const float sq_norm = block_sum(ss, red);
  const float c1 = sq_norm + 1.0f;

  // Pass 2: distances (kept in registers) and their block-wide minimum
  // (= max of -dist, the logsumexp stabilizer).
  float d[ELEMS];
  float dmin = 3.0e38f;
#pragma unroll
  for (int k = 0; k < CHUNKS; ++k) {
    const float4 v = vals[k];
    d[k * 4 + 0] = __builtin_sqrtf(fmaxf(c1 - 2.0f * v.x, 0.0f));
    d[k * 4 + 1] = __builtin_sqrtf(fmaxf(c1 - 2.0f * v.y, 0.0f));
    d[k * 4 + 2] = __builtin_sqrtf(fmaxf(c1 - 2.0f * v.z, 0.0f));
    d[k * 4 + 3] = __builtin_sqrtf(fmaxf(c1 - 2.0f * v.w, 0.0f));
#pragma unroll
    for (int j = 0; j < 4; ++j) dmin = fminf(dmin, d[k * 4 + j]);
  }
  const float dm = block_min(dmin, red);

  // Pass 3: stable sum of exponentials.
  float es = 0.0f;
#pragma unroll
  for (int i = 0; i < ELEMS; ++i) es += __expf(dm - d[i]);
  const float S = block_sum(es, red);

  if (t == 0) row_lse[m] = __logf(S) - dm;   // logsumexp(-dist) for row m
}

// ---- final mean over rows (deterministic single-block tree) ---------------
__global__ void __launch_bounds__(BLOCK)
mean_kernel(const float* __restrict__ row_lse, float* __restrict__ out, int M) {
  __shared__ float red[BLOCK / 32];
  float s = 0.0f;
  for (int i = threadIdx.x; i < M; i += BLOCK) s += row_lse[i];
  s = block_sum(s, red);
  if (threadIdx.x == 0) out[0] = s / (float)M;
}

extern "C" void kernel_launch(void* const* d_in, const int* in_sizes, int n_in,
                              void* d_out, int out_size, void* d_ws, size_t ws_size,
                              hipStream_t stream) {
  const float* x = (const float*)d_in[0];
  const int total = in_sizes[0];
  const int M = total / N_COLS;            // 8192 for the reference shapes

  float* row_lse = (float*)d_ws;           // M floats of scratch (32 KB)

  row_lse_kernel<<<dim3((unsigned)M), dim3(BLOCK), 0, stream>>>(x, row_lse, M);
  mean_kernel<<<dim3(1), dim3(BLOCK), 0, stream>>>(row_lse, (float*)d_out, M);
}